// GL_13322988552756
// MI455X (gfx1250) — compile-verified
//
#include <hip/hip_runtime.h>
#include <hip/hip_bf16.h>
#include <math.h>

// ---------------------------------------------------------------------------
// Problem constants (from reference)
// ---------------------------------------------------------------------------
#define B_SZ   32
#define DIM    8192
#define HID    4096
#define PROJ   256
#define NPART  8
#define NVOLS  4

typedef float v2f __attribute__((ext_vector_type(2)));
typedef float v8f __attribute__((ext_vector_type(8)));

// V_WMMA_F32_16X16X4_F32 : D[16x16] = A[16x4] * B[4x16] + C
__device__ __forceinline__ v8f wmma_f32_16x16x4(v2f a, v2f b, v8f c) {
    return __builtin_amdgcn_wmma_f32_16x16x4_f32(
        false, a, false, b, (short)0, c, false, false);
}

// ---------------------------------------------------------------------------
// Kernel 1: h = x @ W1 + b1 ; BN(train stats over batch) ; ReLU -> h_bn
// Grid: 256 blocks (one per 16-col tile of HID), block = 256 thr = 8 waves.
// Each wave: full 32(M) x 16(N) tile over a K/8 slice; LDS split-K reduce.
// Block owns entire BN columns -> batch stats computed locally, no atomics.
// W1 (128 MB) is streamed exactly once device-wide -> HBM-bound, ~5.7 us.
// ---------------------------------------------------------------------------
__global__ void __launch_bounds__(256)
gl_gemm1_bn_relu(const float* __restrict__ x,
                 const float* __restrict__ W1,
                 const float* __restrict__ b1,
                 const float* __restrict__ gamma,
                 const float* __restrict__ beta,
                 float* __restrict__ h_bn) {
    const int tid  = threadIdx.x;
    const int lane = tid & 31;
    const int wave = tid >> 5;          // 0..7
    const int half = lane >> 4;         // 0,1 (K sub-row pair select)
    const int lid  = lane & 15;         // M row (A) / N col (B)
    const int n0   = blockIdx.x * 16;   // output column tile base

    v8f c_lo = {0.f,0.f,0.f,0.f,0.f,0.f,0.f,0.f}; // rows M=0..15
    v8f c_hi = {0.f,0.f,0.f,0.f,0.f,0.f,0.f,0.f}; // rows M=16..31

    // Per-lane base pointers for this wave's K slice (kk = k_base + 2*half).
    const int kk0 = wave * (DIM / 8) + 2 * half;
    const float* __restrict__ ap_lo = x + (size_t)lid * DIM + kk0;
    const float* __restrict__ ap_hi = x + (size_t)(16 + lid) * DIM + kk0;
    const float* __restrict__ wp    = W1 + (size_t)kk0 * HID + n0 + lid;

    // Compile-time-constant trip count: keeps the loop wave-uniform (no
    // per-lane exit compare, EXEC structurally all-ones for every WMMA).
    #pragma unroll 8
    for (int it = 0; it < (DIM / 8) / 4; ++it) {
        // A tiles: per-lane float2 -> global_load_b64; x is L2-resident (1 MB)
        v2f a_lo = *(const v2f*)ap_lo;
        v2f a_hi = *(const v2f*)ap_hi;
        // B tile: W1 rows kk, kk+1 at cols n0..n0+15
        v2f b;
        b.x = wp[0];
        b.y = wp[HID];
        // Prefetch BOTH row parities of the W1 stream 64 rows (16 iters) ahead
        __builtin_prefetch(wp + (size_t)64 * HID, 0, 1);
        __builtin_prefetch(wp + (size_t)65 * HID, 0, 1);
        c_lo = wmma_f32_16x16x4(a_lo, b, c_lo);
        c_hi = wmma_f32_16x16x4(a_hi, b, c_hi);
        ap_lo += 4; ap_hi += 4;
        wp += (size_t)4 * HID;
    }

    // --- split-K reduction across the 8 waves via LDS ---
    __shared__ float part[8][32][16];   // 16 KB
    __shared__ float hsum[32][16];      // 2 KB
    __shared__ float s_mu[16], s_rs[16];

    #pragma unroll
    for (int v = 0; v < 8; ++v) {
        part[wave][v + 8 * half][lid]      = c_lo[v];
        part[wave][16 + v + 8 * half][lid] = c_hi[v];
    }
    __syncthreads();

    for (int e = tid; e < 512; e += 256) {
        const int m = e >> 4, n = e & 15;
        float s = 0.f;
        #pragma unroll
        for (int w = 0; w < 8; ++w) s += part[w][m][n];
        hsum[m][n] = s + b1[n0 + n];    // b1 cancels in BN, kept for fidelity
    }
    __syncthreads();

    // --- BN training-mode column stats (mean / biased var over batch=32) ---
    if (tid < 16) {
        float mu = 0.f, sq = 0.f;
        #pragma unroll
        for (int m = 0; m < 32; ++m) {
            const float v = hsum[m][tid];
            mu += v; sq += v * v;
        }
        mu *= (1.f / 32.f);
        const float var = sq * (1.f / 32.f) - mu * mu;
        s_mu[tid] = mu;
        s_rs[tid] = rsqrtf(var + 1e-5f);
    }
    __syncthreads();

    for (int e = tid; e < 512; e += 256) {
        const int m = e >> 4, n = e & 15;
        const float v = gamma[n0 + n] * (hsum[m][n] - s_mu[n]) * s_rs[n]
                        + beta[n0 + n];
        h_bn[(size_t)m * HID + n0 + n] = fmaxf(v, 0.f);  // ReLU
    }
}

// ---------------------------------------------------------------------------
// Kernel 2: proj = h_bn @ W2 + b2   [32, 256]
// Grid: 16 blocks (one per 16-col tile), block = 256 thr = 8 waves, split-K.
// Only ~4.5 MB of traffic -> latency-bound; same structure as kernel 1.
// ---------------------------------------------------------------------------
__global__ void __launch_bounds__(256)
gl_gemm2(const float* __restrict__ h_bn,
         const float* __restrict__ W2,
         const float* __restrict__ b2,
         float* __restrict__ proj) {
    const int tid  = threadIdx.x;
    const int lane = tid & 31;
    const int wave = tid >> 5;
    const int half = lane >> 4;
    const int lid  = lane & 15;
    const int n0   = blockIdx.x * 16;

    v8f c_lo = {0.f,0.f,0.f,0.f,0.f,0.f,0.f,0.f};
    v8f c_hi = {0.f,0.f,0.f,0.f,0.f,0.f,0.f,0.f};

    const int kk0 = wave * (HID / 8) + 2 * half;
    const float* __restrict__ ap_lo = h_bn + (size_t)lid * HID + kk0;
    const float* __restrict__ ap_hi = h_bn + (size_t)(16 + lid) * HID + kk0;
    const float* __restrict__ wp    = W2 + (size_t)kk0 * PROJ + n0 + lid;

    #pragma unroll 8
    for (int it = 0; it < (HID / 8) / 4; ++it) {
        v2f a_lo = *(const v2f*)ap_lo;
        v2f a_hi = *(const v2f*)ap_hi;
        v2f b;
        b.x = wp[0];
        b.y = wp[PROJ];
        c_lo = wmma_f32_16x16x4(a_lo, b, c_lo);
        c_hi = wmma_f32_16x16x4(a_hi, b, c_hi);
        ap_lo += 4; ap_hi += 4;
        wp += (size_t)4 * PROJ;
    }

    __shared__ float part[8][32][16];
    #pragma unroll
    for (int v = 0; v < 8; ++v) {
        part[wave][v + 8 * half][lid]      = c_lo[v];
        part[wave][16 + v + 8 * half][lid] = c_hi[v];
    }
    __syncthreads();

    for (int e = tid; e < 512; e += 256) {
        const int m = e >> 4, n = e & 15;
        float s = 0.f;
        #pragma unroll
        for (int w = 0; w < 8; ++w) s += part[w][m][n];
        proj[(size_t)m * PROJ + n0 + n] = s + b2[n0 + n];
    }
}

// ---------------------------------------------------------------------------
// Kernel 3: contrastive loss on proj[32,256] -> scalar. One block, 1024 thr
// (= exactly the 32x32 (i,j) pair grid), wave32 shuffle reductions.
// ---------------------------------------------------------------------------
__global__ void __launch_bounds__(1024)
gl_loss(const float* __restrict__ proj, float* __restrict__ out) {
    __shared__ float p[32][PROJ];   // 32 KB
    __shared__ float E[32][32];
    __shared__ float rn[32], cn[32], nrm[32], wsum[32];

    const int tid  = threadIdx.x;
    const int lane = tid & 31;
    const int wave = tid >> 5;

    for (int e = tid; e < 32 * PROJ; e += 1024) p[e >> 8][e & 255] = proj[e];
    __syncthreads();

    // row L2 norms: wave w owns row w
    float s = 0.f;
    for (int j = lane; j < PROJ; j += 32) { const float v = p[wave][j]; s += v * v; }
    #pragma unroll
    for (int off = 16; off > 0; off >>= 1) s += __shfl_xor(s, off, 32);
    if (lane == 0) nrm[wave] = rsqrtf(s);
    __syncthreads();

    for (int e = tid; e < 32 * PROJ; e += 1024) p[e >> 8][e & 255] *= nrm[e >> 8];
    __syncthreads();

    // E[i][j] = exp(dot(p_i, p_j) / TEMP), raw exp exactly like reference
    {
        const int i = tid >> 5, j = tid & 31;
        float d = 0.f;
        for (int k = 0; k < PROJ; ++k) d = fmaf(p[i][k], p[j][k], d);
        E[i][j] = expf(d * 10.0f);      // 1/TEMP = 10
    }
    __syncthreads();

    // row_neg[i] = sum_{j: i%8 != j%8} E[i][j]; col_neg[i] same on E^T
    {
        const int i = wave, j = lane;
        const bool pos = ((i & 7) == (j & 7));
        float v = pos ? 0.f : E[i][j];
        float u = pos ? 0.f : E[j][i];
        #pragma unroll
        for (int off = 16; off > 0; off >>= 1) {
            v += __shfl_xor(v, off, 32);
            u += __shfl_xor(u, off, 32);
        }
        if (lane == 0) { rn[i] = v; cn[i] = u; }
    }
    __syncthreads();

    // per-(i, pos j) terms: t1 + t2 ; final scale 1/(NVOLS * 2B) = 1/256
    float term = 0.f;
    {
        const int i = tid >> 5, j = tid & 31;
        if ((i & 7) == (j & 7)) {
            const float e1 = E[i][j];   // row direction
            const float e2 = E[j][i];   // column direction (E^T)
            term = logf((e1 + rn[i]) / e1) + logf((e2 + cn[i]) / e2);
        }
    }
    #pragma unroll
    for (int off = 16; off > 0; off >>= 1) term += __shfl_xor(term, off, 32);
    if (lane == 0) wsum[wave] = term;
    __syncthreads();
    if (wave == 0) {
        float t = wsum[lane];
        #pragma unroll
        for (int off = 16; off > 0; off >>= 1) t += __shfl_xor(t, off, 32);
        if (lane == 0) out[0] = t * (1.0f / 256.0f);
    }
}

// ---------------------------------------------------------------------------
// Host launcher
// ---------------------------------------------------------------------------
extern "C" void kernel_launch(void* const* d_in, const int* in_sizes, int n_in,
                              void* d_out, int out_size, void* d_ws, size_t ws_size,
                              hipStream_t stream) {
    const float* x     = (const float*)d_in[0];
    const float* W1    = (const float*)d_in[1];
    const float* b1    = (const float*)d_in[2];
    const float* gamma = (const float*)d_in[3];
    const float* beta  = (const float*)d_in[4];
    const float* W2    = (const float*)d_in[5];
    const float* b2    = (const float*)d_in[6];
    float* out = (float*)d_out;

    float* h_bn = (float*)d_ws;                            // 32*4096 f32 = 512 KB
    float* proj = h_bn + (size_t)B_SZ * HID;               // 32*256  f32 =  32 KB

    gl_gemm1_bn_relu<<<HID / 16, 256, 0, stream>>>(x, W1, b1, gamma, beta, h_bn);
    gl_gemm2<<<PROJ / 16, 256, 0, stream>>>(h_bn, W2, b2, proj);
    gl_loss<<<1, 1024, 0, stream>>>(proj, out);
}